// RouterOursNoNew_27788438405470
// MI455X (gfx1250) — compile-verified
//
#include <hip/hip_runtime.h>
#include <hip/hip_bf16.h>
#include <math.h>

// ---------------------------------------------------------------------------
// RouterOursNoNew for MI455X (gfx1250, wave32).
// Phase 1: stream 805MB of scores once; masked q-reduction via
//          V_WMMA_F32_16X16X4_F32 (A = ones, B = mask-premultiplied 4x16 tile).
// Phase 2: per-batch bitonic sort in LDS -> threshold -> stable prefix-scan
//          selection (reproduces sorted top_k indices), gather.
// ---------------------------------------------------------------------------

typedef __attribute__((ext_vector_type(2))) float v2f;
typedef __attribute__((ext_vector_type(8))) float v8f;

#define B_ 4
#define H_ 12
#define L_ 2048
#define D_ 768
#define QSPLIT 8
#define QLEN (L_ / QSPLIT)        // 256 q-rows per block
#define COLS_PER_BLOCK 64         // 4 waves x 16 columns

// ---------------------------------------------------------------------------
// Kernel 1: importance partials. grid = (L/64, QSPLIT, B), block = 128 (4 waves).
// Each wave owns 16 columns; iterates h (12) and q in chunks of 4.
// Per chunk: two coalesced global_load_b32 (rows q0+krow, q0+krow+1 for this
// lane-half), premultiply by mask_f[q], one wmma accumulate into v8f.
// Lane<16 of acc[0] holds the 16 column sums (D row M=0).
// ---------------------------------------------------------------------------
__global__ __launch_bounds__(128) void reduce_scores(
    const float* __restrict__ scores,   // (B,H,L,L)
    const float* __restrict__ attn,     // (B,1,1,L)
    float* __restrict__ imp_part)       // (QSPLIT, B, L)
{
  const int colblk = blockIdx.x;        // 0..31
  const int qblk   = blockIdx.y;        // 0..QSPLIT-1
  const int b      = blockIdx.z;
  const int tid    = threadIdx.x;
  const int wave   = tid >> 5;
  const int lane   = tid & 31;

  __shared__ float smask[QLEN];
  for (int i = tid; i < QLEN; i += 128) {
    float mv = attn[b * L_ + qblk * QLEN + i];
    smask[i] = (mv > -10.0f) ? 1.0f : 0.0f;
  }
  __syncthreads();

  const int l0   = colblk * COLS_PER_BLOCK + wave * 16;
  const int col  = l0 + (lane & 15);
  const int krow = (lane < 16) ? 0 : 2; // which K rows this lane-half carries

  v2f a; a.x = 1.0f; a.y = 1.0f;        // A = all-ones 16x4 (K-perm invariant)
  v8f acc = {};

  for (int h = 0; h < H_; ++h) {
    const float* base =
        scores + (((size_t)b * H_ + h) * L_ + (size_t)qblk * QLEN) * L_;
    const float* p = base + (size_t)krow * L_ + col;
#pragma unroll 4
    for (int qc = 0; qc < QLEN; qc += 4) {
      float m0 = smask[qc + krow];
      float m1 = smask[qc + krow + 1];
      v2f bm;
      bm.x = p[0]  * m0;                // scores[q0+krow  , col] * mask
      bm.y = p[L_] * m1;                // scores[q0+krow+1, col] * mask
      acc = __builtin_amdgcn_wmma_f32_16x16x4_f32(
          false, a, false, bm, (short)0, acc, false, false);
      p += 4 * L_;
    }
  }

  if (lane < 16) {
    // single writer per (qblk, b, col): deterministic, no atomics
    imp_part[(size_t)qblk * (B_ * L_) + (size_t)b * L_ + col] = acc[0];
  }
}

// ---------------------------------------------------------------------------
// Kernel 2: per-batch top-K with stable tie handling. block = 1024 threads.
// ---------------------------------------------------------------------------
__global__ __launch_bounds__(1024) void topk_select(
    const float* __restrict__ imp_part, // (QSPLIT, B, L)
    const float* __restrict__ attn,     // (B,1,1,L)
    float* __restrict__ out,            // flat outputs
    int* __restrict__ idxbuf,           // (B, K)
    int K)
{
  const int b = blockIdx.x;
  const int t = threadIdx.x;
  const float scale = 1.0f / (float)(H_ * L_);

  __shared__ float    vals[L_];
  __shared__ float    sortb[L_];
  __shared__ unsigned scanA[L_];
  __shared__ unsigned scanB[L_];

  for (int i = t; i < L_; i += 1024) {
    float s = 0.0f;
    for (int qs = 0; qs < QSPLIT; ++qs)         // fixed order -> deterministic
      s += imp_part[(size_t)qs * (B_ * L_) + (size_t)b * L_ + i];
    float mv = attn[b * L_ + i];
    float mf = (mv > -10.0f) ? 1.0f : 0.0f;
    float v = s * mf * scale;
    if (i == 0) v = INFINITY;                   // CLS always kept
    vals[i] = v;
    sortb[i] = v;
  }
  __syncthreads();

  // Bitonic sort, descending.
  for (int k = 2; k <= L_; k <<= 1) {
    for (int j = k >> 1; j > 0; j >>= 1) {
      for (int i = t; i < L_; i += 1024) {
        int ixj = i ^ j;
        if (ixj > i) {
          float x = sortb[i], y = sortb[ixj];
          bool descBlock = ((i & k) == 0);
          if (descBlock ? (x < y) : (x > y)) { sortb[i] = y; sortb[ixj] = x; }
        }
      }
      __syncthreads();
    }
  }
  const float thresh = sortb[K - 1];            // K-th largest

  // Packed flags: (greater << 16) | equal ; inclusive Hillis-Steele scan.
  for (int i = t; i < L_; i += 1024) {
    float v = vals[i];
    unsigned g = (v > thresh) ? 1u : 0u;
    unsigned e = (v == thresh) ? 1u : 0u;
    scanA[i] = (g << 16) | e;
  }
  __syncthreads();
  unsigned* src = scanA;
  unsigned* dst = scanB;
  for (int d = 1; d < L_; d <<= 1) {
    for (int i = t; i < L_; i += 1024)
      dst[i] = src[i] + ((i >= d) ? src[i - d] : 0u);
    __syncthreads();
    unsigned* tmp = src; src = dst; dst = tmp;
  }
  const unsigned total = src[L_ - 1];
  const int cntG = (int)(total >> 16);
  const int ties = K - cntG;                    // >= 1 by construction

  const size_t maskOff = (size_t)B_ * K * D_;
  const size_t tomeOff = maskOff + (size_t)B_ * K;

  // Stable selection: all > thresh, plus first `ties` equals in index order.
  for (int i = t; i < L_; i += 1024) {
    float v = vals[i];
    unsigned incl = src[i];
    int g = (v > thresh) ? 1 : 0;
    int e = (v == thresh) ? 1 : 0;
    int pG = (int)(incl >> 16) - g;             // exclusive prefixes
    int pE = (int)(incl & 0xffffu) - e;
    bool sel = g || (e && pE < ties);
    if (sel) {
      int pos = pG + (pE < ties ? pE : ties);   // output slot, index-sorted
      if (pos < K) {
        idxbuf[b * K + pos] = i;
        out[maskOff + (size_t)b * K + pos] = attn[b * L_ + i];
      }
    }
  }
  for (int kk = t; kk < K; kk += 1024)
    out[tomeOff + (size_t)b * K + kk] = 1.0f;
}

// ---------------------------------------------------------------------------
// Kernel 3: gather preserved tokens. grid = B*K blocks, 192 thr, float4 rows.
// ---------------------------------------------------------------------------
__global__ __launch_bounds__(192) void gather_tokens(
    const float* __restrict__ hidden,   // (B, L, D)
    const int* __restrict__ idxbuf,     // (B, K)
    float* __restrict__ out,            // tokens at offset 0
    int K)
{
  const int x = blockIdx.x;             // b*K + k
  const int b = x / K;
  const int row = idxbuf[x];
  const float4* src =
      (const float4*)(hidden + ((size_t)b * L_ + row) * D_);
  float4* dst = (float4*)(out + (size_t)x * D_);
  dst[threadIdx.x] = src[threadIdx.x];
}

extern "C" void kernel_launch(void* const* d_in, const int* in_sizes, int n_in,
                              void* d_out, int out_size, void* d_ws, size_t ws_size,
                              hipStream_t stream) {
  (void)in_sizes; (void)n_in; (void)ws_size;

  const float* hidden = (const float*)d_in[0];
  const float* attn   = (const float*)d_in[1];
  const float* scores = (const float*)d_in[2];

  // out_size = B*K*D + B*K + B*K = B*K*(D+2)  ->  K (host-side, capture-safe)
  int K = out_size / (B_ * (D_ + 2));
  if (K > L_) K = L_;
  if (K < 1)  K = 1;

  float* imp_part = (float*)d_ws;                                  // 256 KB
  int*   idxbuf   = (int*)((char*)d_ws + (size_t)QSPLIT * B_ * L_ * sizeof(float));

  float* out = (float*)d_out;

  reduce_scores<<<dim3(L_ / COLS_PER_BLOCK, QSPLIT, B_), 128, 0, stream>>>(
      scores, attn, imp_part);
  topk_select<<<dim3(B_), 1024, 0, stream>>>(imp_part, attn, out, idxbuf, K);
  gather_tokens<<<dim3(B_ * K), 192, 0, stream>>>(hidden, idxbuf, out, K);
}